// FullAttention_38697655337351
// MI455X (gfx1250) — compile-verified
//
#include <hip/hip_runtime.h>

typedef float v2f __attribute__((ext_vector_type(2)));
typedef float v8f __attribute__((ext_vector_type(8)));

namespace {
constexpr int Lc = 768, Hc = 8, Ec = 64;
constexpr int TOK_STRIDE = Hc * Ec;          // 512 floats between tokens
constexpr int BV_STRIDE  = Lc * TOK_STRIDE;  // per (b,v) slab
constexpr int NBLK = 128;                    // 127 interior + 1 boundary per (b,v,h)
constexpr float SCALE = 0.125f;              // 1/sqrt(64)

// Row r (0..5) of attention-block pblk -> token index in L.
__device__ __forceinline__ int tok_of(int r, int pblk) {
  // boundary block: rows 0-2 -> tokens 0-2, rows 3-5 -> tokens 765-767
  return (pblk == NBLK - 1) ? ((r < 3) ? r : r + (Lc - 6)) : (3 + 6 * pblk + r);
}

__device__ __forceinline__ float bperm(int srcByte, float x) {
  return __int_as_float(__builtin_amdgcn_ds_bpermute(srcByte, __float_as_int(x)));
}
} // namespace

__global__ __launch_bounds__(256) void
neighbour_attn_wmma(const float* __restrict__ Q, const float* __restrict__ K,
                    const float* __restrict__ V, float* __restrict__ O) {
  const int lane = threadIdx.x & 31;
  const int wave = threadIdx.x >> 5;
  const int gidx = blockIdx.x * 8 + wave;     // 0..65535 attention blocks
  const int pblk = gidx & (NBLK - 1);
  const int h    = (gidx >> 7) & (Hc - 1);
  const int bv   = gidx >> 10;                // 0..63
  const int base = bv * BV_STRIDE + h * Ec;

  const int lm = lane & 15;   // row index (A) / column index (B,C,D)
  const int hs = lane >> 4;   // which K-pair this half-wave supplies

  // Clamp rows >= 6 to row 0: loads stay in-bounds and finite; the junk only
  // reaches score rows/cols >= 6 and D rows >= 6, none of which are used.
  const int frow   = (lm < 6) ? lm : 0;
  const int rowoff = base + tok_of(frow, pblk) * TOK_STRIDE + 2 * hs;

  // ---------- scores^T = K * Q^T   (M = key row s, N = query col l, K = E) ----------
  // f32 A-frag layout == f32 B-frag layout under row-major [token][feature]:
  //   lane l: row = l&15, features e = 4j + 2*(l>>4) + {0,1}
  v8f c = {};
#pragma unroll
  for (int j = 0; j < 16; ++j) {
    const v2f kf = *(const v2f*)(K + rowoff + 4 * j);   // A: keys   (M=s, K=e)
    const v2f qf = *(const v2f*)(Q + rowoff + 4 * j);   // B: Q^T    (K=e, N=l)
    c = __builtin_amdgcn_wmma_f32_16x16x4_f32(false, kf, false, qf, (short)0, c,
                                              false, false);
  }

  // ---------- softmax: lane l (<16) owns all 6 scores of query l in c[0..5] ----------
  const float m =
      fmaxf(fmaxf(fmaxf(c[0], c[1]), fmaxf(c[2], c[3])), fmaxf(c[4], c[5]));
  float p0 = __expf((c[0] - m) * SCALE);
  float p1 = __expf((c[1] - m) * SCALE);
  float p2 = __expf((c[2] - m) * SCALE);
  float p3 = __expf((c[3] - m) * SCALE);
  float p4 = __expf((c[4] - m) * SCALE);
  float p5 = __expf((c[5] - m) * SCALE);
  const float inv = 1.0f / (p0 + p1 + p2 + p3 + p4 + p5);
  p0 *= inv; p1 *= inv; p2 *= inv; p3 *= inv; p4 *= inv; p5 *= inv;

  // ---------- A-fragments of P for O = P * V  (M = query, K = s) ----------
  // Lane l needs P[row = l&15][s = 4*kc + 2*hs + {0,1}]; P[row] lives in lane row.
  const int src = lm << 2;
  const float g0 = bperm(src, p0), g1 = bperm(src, p1), g2 = bperm(src, p2);
  const float g3 = bperm(src, p3), g4 = bperm(src, p4), g5 = bperm(src, p5);
  v2f a0, a1;
  a0.x = hs ? g2 : g0;  a0.y = hs ? g3 : g1;      // kc=0: s = {0,1} / {2,3}
  a1.x = hs ? 0.f : g4; a1.y = hs ? 0.f : g5;     // kc=1: s = {4,5} / {6,7}=0

  // V row bases (B operand rows s). kc=1/hi-half rows 6,7 are clamped (A is 0 there).
  const int vb0 = base + tok_of(2 * hs, pblk) * TOK_STRIDE;
  const int vb1 = base + tok_of(2 * hs + 1, pblk) * TOK_STRIDE;
  const int vb2 = base + tok_of(hs ? 0 : 4, pblk) * TOK_STRIDE;
  const int vb3 = base + tok_of(hs ? 0 : 5, pblk) * TOK_STRIDE;

  // ---------- O = P * V over 4 column tiles of E ----------
#pragma unroll
  for (int nt = 0; nt < 4; ++nt) {
    const int col = nt * 16 + lm;
    v2f b0, b1;                                  // B: V[s][e]
    b0.x = V[vb0 + col]; b0.y = V[vb1 + col];
    b1.x = V[vb2 + col]; b1.y = V[vb3 + col];
    v8f d = {};
    d = __builtin_amdgcn_wmma_f32_16x16x4_f32(false, a0, false, b0, (short)0, d,
                                              false, false);
    d = __builtin_amdgcn_wmma_f32_16x16x4_f32(false, a1, false, b1, (short)0, d,
                                              false, false);
    if (lane < 16) {                             // D rows 0..5 live in lanes 0-15
#pragma unroll
      for (int r = 0; r < 6; ++r) {
        O[base + tok_of(r, pblk) * TOK_STRIDE + col] = d[r];
      }
    }
  }
}

extern "C" void kernel_launch(void* const* d_in, const int* in_sizes, int n_in,
                              void* d_out, int out_size, void* d_ws, size_t ws_size,
                              hipStream_t stream) {
  const float* Q = (const float*)d_in[0];
  const float* K = (const float*)d_in[1];
  const float* V = (const float*)d_in[2];
  float* O = (float*)d_out;
  // 65536 attention blocks, 8 waves (one block each) per 256-thread workgroup.
  neighbour_attn_wmma<<<dim3(8192), dim3(256), 0, stream>>>(Q, K, V, O);
}